// TimeAwareDyGraph_82154134438717
// MI455X (gfx1250) — compile-verified
//
#include <hip/hip_runtime.h>
#include <hip/hip_bf16.h>

// ---------------------------------------------------------------------------
// CDNA5 (gfx1250) implementation using V_WMMA_F32_16X16X4_F32 for all dense
// layers (full f32 precision). Wave32; each wave owns a 16-row tile.
// Weights are staged zero-padded in LDS once per block so the WMMA inner
// loops are pure ds_load + v_wmma (no exec-mask branches).
// ---------------------------------------------------------------------------

typedef float v2f __attribute__((ext_vector_type(2)));
typedef float v8f __attribute__((ext_vector_type(8)));

#define NCENT 100000
#define TSEG 3
#define DDIM 20
#define SS 512
#define UU 1024

// wave-local LDS phase fence: compiler barrier + wait for all DS ops
#define DSFENCE() asm volatile("s_wait_dscnt 0x0" ::: "memory")

__device__ __forceinline__ v8f wmma4(v2f a, v2f b, v8f c) {
  // D(16x16,f32) = A(16x4,f32) x B(4x16,f32) + C
  return __builtin_amdgcn_wmma_f32_16x16x4_f32(false, a, false, b, (short)0, c,
                                               false, false);
}

// A fragment (16x4 f32): lane m (m<16) holds K=k0,k0+1 ; lane m+16 holds K=k0+2,k0+3
__device__ __forceinline__ v2f load_a_lds(const float* in, int stride, int k0,
                                          int lane) {
  int m = lane & 15;
  int kb = k0 + ((lane >> 4) << 1);
  const float* p = in + m * stride + kb;
  v2f a;
  a[0] = p[0];
  a[1] = p[1];
  return a;
}

// B fragment (4x16 f32) from LDS-staged zero-padded weights Wp[K][ncpad]
__device__ __forceinline__ v2f load_b_lds(const float* Wp, int ncpad, int k0,
                                          int n0, int lane) {
  int n = n0 + (lane & 15);
  int kb = k0 + ((lane >> 4) << 1);
  const float* p = Wp + kb * ncpad + n;
  v2f b;
  b[0] = p[0];
  b[1] = p[ncpad];
  return b;
}

// C/D fragment: lane holds col n = n0+(lane&15); VGPR r holds row r + 8*(lane>>4)
__device__ __forceinline__ v8f bias_relu(v8f c, const float* biasP, int n0,
                                         int lane) {
  float bv = biasP[n0 + (lane & 15)];  // zero-padded bias
#pragma unroll
  for (int r = 0; r < 8; ++r) c[r] = fmaxf(c[r] + bv, 0.f);
  return c;
}

// unguarded C store into padded LDS tile (padding cols never read)
__device__ __forceinline__ void store_c_lds(v8f c, float* out, int stride,
                                            int n0, int lane) {
  int n = n0 + (lane & 15);
  int mrow = (lane >> 4) << 3;
#pragma unroll
  for (int r = 0; r < 8; ++r) out[(mrow + r) * stride + n] = c[r];
}

// ---------------------------------------------------------------------------
// Kernel 1: vec_out[T*N][10] = relu(vecs @ Wv1 + bv1) @ Wv2 + bv2
// 16 rows / wave, 4 waves / block.
// ---------------------------------------------------------------------------
__global__ __launch_bounds__(128) void vecout_kernel(
    const float* __restrict__ vecs, const float* __restrict__ Wv1,
    const float* __restrict__ bv1, const float* __restrict__ Wv2,
    const float* __restrict__ bv2, float* __restrict__ vec_out, int R) {
  __shared__ float sWv1[20 * 32];
  __shared__ float sWv2[20 * 16];
  __shared__ float sbv1[32];
  __shared__ float sbv2[16];
  __shared__ float s_in[4][16 * 20];
  __shared__ float s_h[4][16 * 32];

  int tid = threadIdx.x;
  int lane = tid & 31;
  int wid = tid >> 5;

  // stage zero-padded weights once per block
  for (int i = tid; i < 20 * 32; i += 128) {
    int k = i >> 5, n = i & 31;
    sWv1[i] = (n < 20) ? Wv1[k * 20 + n] : 0.f;
  }
  for (int i = tid; i < 20 * 16; i += 128) {
    int k = i >> 4, n = i & 15;
    sWv2[i] = (n < 10) ? Wv2[k * 10 + n] : 0.f;
  }
  if (tid < 32) sbv1[tid] = (tid < 20) ? bv1[tid] : 0.f;
  if (tid < 16) sbv2[tid] = (tid < 10) ? bv2[tid] : 0.f;

  int rowbase = (blockIdx.x * 4 + wid) * 16;
  float* in = s_in[wid];
  float* hh = s_h[wid];

  int e = lane & 15;
  int row = rowbase + e;
  if (row > R - 1) row = R - 1;  // clamp tail (stores guarded below)
  int f0 = (lane >> 4) * 10;
#pragma unroll
  for (int j = 0; j < 10; ++j)
    in[e * 20 + f0 + j] = vecs[(size_t)row * 20 + f0 + j];
  __syncthreads();

  // layer 1: [16,20] @ [20,20], two N-tiles (0..15, 16..31-padded)
  v8f c0 = {};
  v8f c1 = {};
#pragma unroll
  for (int k0 = 0; k0 < 20; k0 += 4) {
    v2f a = load_a_lds(in, 20, k0, lane);
    c0 = wmma4(a, load_b_lds(sWv1, 32, k0, 0, lane), c0);
    c1 = wmma4(a, load_b_lds(sWv1, 32, k0, 16, lane), c1);
  }
  c0 = bias_relu(c0, sbv1, 0, lane);
  c1 = bias_relu(c1, sbv1, 16, lane);
  store_c_lds(c0, hh, 32, 0, lane);
  store_c_lds(c1, hh, 32, 16, lane);
  DSFENCE();

  // layer 2: [16,20] @ [20,10]
  v8f o = {};
#pragma unroll
  for (int k0 = 0; k0 < 20; k0 += 4)
    o = wmma4(load_a_lds(hh, 32, k0, lane), load_b_lds(sWv2, 16, k0, 0, lane),
              o);

  int n = lane & 15;
  if (n < 10) {
    float bv = sbv2[n];
    int mrow = (lane >> 4) << 3;
#pragma unroll
    for (int r = 0; r < 8; ++r) {
      int rr = rowbase + mrow + r;
      if (rr < R) vec_out[(size_t)rr * 10 + n] = o[r] + bv;
    }
  }
}

// ---------------------------------------------------------------------------
// Kernel 2: per-element loss terms. 16 elements / wave, 4 waves / block.
// ---------------------------------------------------------------------------
__global__ __launch_bounds__(128) void main_kernel(
    const int* __restrict__ x, const int* __restrict__ xseg,
    const int* __restrict__ y, const int* __restrict__ yseg,
    const int* __restrict__ neg_idx, const float* __restrict__ vecs,
    const float* __restrict__ time_emb, const float* __restrict__ W1,
    const float* __restrict__ b1, const float* __restrict__ W2,
    const float* __restrict__ b2, const float* __restrict__ Wv1,
    const float* __restrict__ bv1, const float* __restrict__ Wv2,
    const float* __restrict__ bv2, const float* __restrict__ vec_out,
    float* __restrict__ partials) {
  __shared__ float sW1[60 * 32];
  __shared__ float sW2[20 * 16];
  __shared__ float sWv1[20 * 32];
  __shared__ float sWv2[20 * 16];
  __shared__ float sb1[32];
  __shared__ float sb2[16];
  __shared__ float sbv1[32];
  __shared__ float sbv2[16];
  __shared__ float s_xin[4][16 * 60];
  __shared__ float s_ye[4][16 * 20];
  __shared__ float s_h[4][16 * 32];
  __shared__ float s_xi[4][16 * 16];
  __shared__ float s_pos[4][16 * 16];
  __shared__ float s_pd[4][16];
  __shared__ float s_red[128];

  int tid = threadIdx.x;
  int lane = tid & 31;
  int wid = tid >> 5;

  // stage zero-padded weights/biases once per block
  for (int i = tid; i < 60 * 32; i += 128) {
    int k = i >> 5, n = i & 31;
    sW1[i] = (n < 20) ? W1[k * 20 + n] : 0.f;
  }
  for (int i = tid; i < 20 * 32; i += 128) {
    int k = i >> 5, n = i & 31;
    sWv1[i] = (n < 20) ? Wv1[k * 20 + n] : 0.f;
  }
  for (int i = tid; i < 20 * 16; i += 128) {
    int k = i >> 4, n = i & 15;
    sW2[i] = (n < 10) ? W2[k * 10 + n] : 0.f;
    sWv2[i] = (n < 10) ? Wv2[k * 10 + n] : 0.f;
  }
  if (tid < 32) {
    sb1[tid] = (tid < 20) ? b1[tid] : 0.f;
    sbv1[tid] = (tid < 20) ? bv1[tid] : 0.f;
  }
  if (tid < 16) {
    sb2[tid] = (tid < 10) ? b2[tid] : 0.f;
    sbv2[tid] = (tid < 10) ? bv2[tid] : 0.f;
  }

  int base = (blockIdx.x * 4 + wid) * 16;
  int e = lane & 15;
  int gid = base + e;

  int xv = x[gid], xs = xseg[gid], yv = y[gid], ys = yseg[gid];

  float* xin = s_xin[wid];
  float* ye = s_ye[wid];
  float* hh = s_h[wid];
  float* xi = s_xi[wid];
  float* ps = s_pos[wid];
  float* pd = s_pd[wid];

  const float* xrow = vecs + ((size_t)xs * NCENT + xv) * DDIM;
  const float* yrow = vecs + ((size_t)ys * NCENT + yv) * DDIM;
  const float* tex = time_emb + xs * DDIM;
  const float* tey = time_emb + ys * DDIM;

  // xi_in row = [x_emb(20) | time_emb[xs](20) | time_emb[ys](20)]
  int f0 = (lane >> 4) * 30;
  for (int j = 0; j < 30; ++j) {
    int f = f0 + j;
    float v = (f < 20) ? xrow[f] : ((f < 40) ? tex[f - 20] : tey[f - 40]);
    xin[e * 60 + f] = v;
  }
  int g0 = (lane >> 4) * 10;
  for (int j = 0; j < 10; ++j) ye[e * 20 + g0 + j] = yrow[g0 + j];
  __syncthreads();  // weights visible block-wide; own-wave LDS also ordered

  // xi layer 1: [16,60] @ W1[60,20]
  v8f c0 = {};
  v8f c1 = {};
  for (int k0 = 0; k0 < 60; k0 += 4) {
    v2f a = load_a_lds(xin, 60, k0, lane);
    c0 = wmma4(a, load_b_lds(sW1, 32, k0, 0, lane), c0);
    c1 = wmma4(a, load_b_lds(sW1, 32, k0, 16, lane), c1);
  }
  c0 = bias_relu(c0, sb1, 0, lane);
  c1 = bias_relu(c1, sb1, 16, lane);
  store_c_lds(c0, hh, 32, 0, lane);
  store_c_lds(c1, hh, 32, 16, lane);
  DSFENCE();

  // xi layer 2: [16,20] @ W2[20,10] -> xi tile (padded cols 10..15 unused)
  v8f o = {};
#pragma unroll
  for (int k0 = 0; k0 < 20; k0 += 4)
    o = wmma4(load_a_lds(hh, 32, k0, lane), load_b_lds(sW2, 16, k0, 0, lane),
              o);
  {
    float bv = sb2[lane & 15];
    int n = lane & 15;
    int mrow = (lane >> 4) << 3;
#pragma unroll
    for (int r = 0; r < 8; ++r) xi[(mrow + r) * 16 + n] = o[r] + bv;
  }

  // pos layer 1: [16,20] @ Wv1[20,20]  (LDS ops are in-order per wave, so the
  // WAR on hh vs the xi-layer-2 reads is safe)
  v8f d0 = {};
  v8f d1 = {};
#pragma unroll
  for (int k0 = 0; k0 < 20; k0 += 4) {
    v2f a = load_a_lds(ye, 20, k0, lane);
    d0 = wmma4(a, load_b_lds(sWv1, 32, k0, 0, lane), d0);
    d1 = wmma4(a, load_b_lds(sWv1, 32, k0, 16, lane), d1);
  }
  d0 = bias_relu(d0, sbv1, 0, lane);
  d1 = bias_relu(d1, sbv1, 16, lane);
  store_c_lds(d0, hh, 32, 0, lane);
  store_c_lds(d1, hh, 32, 16, lane);
  DSFENCE();

  // pos layer 2: [16,20] @ Wv2[20,10]
  v8f p = {};
#pragma unroll
  for (int k0 = 0; k0 < 20; k0 += 4)
    p = wmma4(load_a_lds(hh, 32, k0, lane), load_b_lds(sWv2, 16, k0, 0, lane),
              p);
  {
    float bv = sbv2[lane & 15];
    int n = lane & 15;
    int mrow = (lane >> 4) << 3;
#pragma unroll
    for (int r = 0; r < 8; ++r) ps[(mrow + r) * 16 + n] = p[r] + bv;
  }
  DSFENCE();

  // pos_d per element
  if (lane < 16) {
    float s = 0.f;
#pragma unroll
    for (int n = 0; n < 10; ++n) {
      float d = xi[lane * 16 + n] - ps[lane * 16 + n];
      s += d * d;
    }
    pd[lane] = sqrtf(s);
  }
  DSFENCE();

  // negatives: 16 elements x 10 negs = 160 tasks, 5 per lane
  float acc = 0.f;
  for (int i = 0; i < 5; ++i) {
    int t = i * 32 + lane;
    int ee = t & 15;
    int k = t >> 4;
    int g = base + ee;
    int idx = neg_idx[g * 10 + k];
    int seg = xseg[g];
    const float* nv = vec_out + ((size_t)seg * NCENT + idx) * 10;
    float s = 0.f;
#pragma unroll
    for (int n = 0; n < 10; ++n) {
      float d = xi[ee * 16 + n] - nv[n];
      s += d * d;
    }
    float nd = sqrtf(s);
    float z = nd - pd[ee];
    // log_sigmoid(z) = min(z,0) - log1p(exp(-|z|))
    acc += fminf(z, 0.f) - log1pf(expf(-fabsf(z)));
  }

  // deterministic block tree-reduce
  s_red[tid] = acc;
  __syncthreads();
  for (int off = 64; off > 0; off >>= 1) {
    if (tid < off) s_red[tid] += s_red[tid + off];
    __syncthreads();
  }
  if (tid == 0) partials[blockIdx.x] = s_red[0];
}

// ---------------------------------------------------------------------------
// Kernel 3: single-block finalize (deterministic fixed-order reduction)
// ---------------------------------------------------------------------------
__device__ __forceinline__ float block_reduce_sum256(float v, float* red) {
  int tid = threadIdx.x;
  red[tid] = v;
  __syncthreads();
  for (int off = 128; off > 0; off >>= 1) {
    if (tid < off) red[tid] += red[tid + off];
    __syncthreads();
  }
  float r = red[0];
  __syncthreads();
  return r;
}

__device__ __forceinline__ float strided_sumsq(const float* p, int n) {
  float s = 0.f;
  for (int i = threadIdx.x; i < n; i += 256) {
    float v = p[i];
    s += v * v;
  }
  return s;
}

__global__ __launch_bounds__(256) void finalize_kernel(
    const float* __restrict__ partials, int nPart,
    const float* time_emb, const float* W1, const float* b1, const float* W2,
    const float* b2, const float* Wv1, const float* bv1, const float* Wv2,
    const float* bv2, const float* Wadj, const float* badj,
    float* __restrict__ out) {
  __shared__ float red[256];
  float reg = 0.f;
  reg += sqrtf(block_reduce_sum256(strided_sumsq(time_emb, TSEG * DDIM), red));
  reg += sqrtf(block_reduce_sum256(strided_sumsq(W1, 60 * 20), red));
  reg += sqrtf(block_reduce_sum256(strided_sumsq(b1, 20), red));
  reg += sqrtf(block_reduce_sum256(strided_sumsq(W2, 20 * 10), red));
  reg += sqrtf(block_reduce_sum256(strided_sumsq(b2, 10), red));
  reg += sqrtf(block_reduce_sum256(strided_sumsq(Wv1, 20 * 20), red));
  reg += sqrtf(block_reduce_sum256(strided_sumsq(bv1, 20), red));
  reg += sqrtf(block_reduce_sum256(strided_sumsq(Wv2, 20 * 10), red));
  reg += sqrtf(block_reduce_sum256(strided_sumsq(bv2, 10), red));
  reg += sqrtf(block_reduce_sum256(strided_sumsq(Wadj, 20 * 10), red));
  reg += sqrtf(block_reduce_sum256(strided_sumsq(badj, 10), red));

  float s = 0.f;
  for (int i = threadIdx.x; i < nPart; i += 256) s += partials[i];
  float total = block_reduce_sum256(s, red);

  if (threadIdx.x == 0)
    out[0] = -total / (float)(SS * UU) + 0.01f * reg;
}

// ---------------------------------------------------------------------------
extern "C" void kernel_launch(void* const* d_in, const int* in_sizes, int n_in,
                              void* d_out, int out_size, void* d_ws,
                              size_t ws_size, hipStream_t stream) {
  const int* x = (const int*)d_in[0];
  const int* xseg = (const int*)d_in[1];
  const int* y = (const int*)d_in[2];
  const int* yseg = (const int*)d_in[3];
  const int* neg_idx = (const int*)d_in[4];
  const float* vecs = (const float*)d_in[5];
  const float* time_emb = (const float*)d_in[6];
  const float* W1 = (const float*)d_in[7];
  const float* b1 = (const float*)d_in[8];
  const float* W2 = (const float*)d_in[9];
  const float* b2 = (const float*)d_in[10];
  const float* Wv1 = (const float*)d_in[11];
  const float* bv1 = (const float*)d_in[12];
  const float* Wv2 = (const float*)d_in[13];
  const float* bv2 = (const float*)d_in[14];
  const float* Wadj = (const float*)d_in[15];
  const float* badj = (const float*)d_in[16];
  float* out = (float*)d_out;

  const int R = TSEG * NCENT;  // 300000 rows
  float* vec_out = (float*)d_ws;
  float* partials = vec_out + (size_t)R * 10;

  vecout_kernel<<<(R + 63) / 64, 128, 0, stream>>>(vecs, Wv1, bv1, Wv2, bv2,
                                                   vec_out, R);

  const int nElem = SS * UU;       // 524288, divisible by 64
  const int nBlocks = nElem / 64;  // 8192 full blocks
  main_kernel<<<nBlocks, 128, 0, stream>>>(x, xseg, y, yseg, neg_idx, vecs,
                                           time_emb, W1, b1, W2, b2, Wv1, bv1,
                                           Wv2, bv2, vec_out, partials);

  finalize_kernel<<<1, 256, 0, stream>>>(partials, nBlocks, time_emb, W1, b1,
                                         W2, b2, Wv1, bv1, Wv2, bv2, Wadj,
                                         badj, out);
}